// InverseLogPolarLayer_46909632807255
// MI455X (gfx1250) — compile-verified
//
#include <hip/hip_runtime.h>
#include <hip/hip_bf16.h>
#include <math.h>

#ifndef M_PI
#define M_PI 3.14159265358979323846
#endif

#define HH 256
#define WW 256
#define CC 512

// Tile: 64 j-pixels x 128 channels per 256-thread workgroup.
#define PIX 64
#define CHT 128
// LDS row stride in floats: 128 + 4 pad => 528B (16B aligned for B128 copies,
// 132 mod 64 = 4 -> only 2-way bank conflict on transposed reads).
#define LDS_STRIDE 132

__global__ __launch_bounds__(256)
void InverseLogPolarLayer_46909632807255_kernel(const float* __restrict__ in,
                                                float* __restrict__ out) {
  __shared__ __align__(16) float tile[PIX * LDS_STRIDE];
  __shared__ int srcbase[PIX];

  const int t  = threadIdx.x;
  const int i  = blockIdx.x;         // output row 0..255
  const int j0 = blockIdx.y * PIX;   // j tile base
  const int c0 = blockIdx.z * CHT;   // channel tile base

  // ---- Phase 0: inverse log-polar indices for this row's 64 pixels.
  // Float64 to match numpy (np.rint = round-half-even = rint under RNE).
  if (t < PIX) {
    const double cy = (double)(HH - 1) * 0.5;  // 127.5
    const double cx = (double)(WW - 1) * 0.5;
    const double dy = (double)i - cy;
    const double dx = (double)(j0 + t) - cx;
    const double r  = sqrt(dx * dx + dy * dy);
    const double rmax = sqrt(cx * cx + cy * cy);
    double rho = (r > 0.5)
                   ? (log(fmax(r, 1e-6)) / log(rmax)) * (double)(WW - 1)
                   : 0.0;
    double th = (atan2(dy, dx) + M_PI) / (2.0 * M_PI) * (double)(HH - 1);
    double rr = rint(rho);
    rr = rr < 0.0 ? 0.0 : (rr > (double)(WW - 1) ? (double)(WW - 1) : rr);
    double tt = rint(th);
    tt = tt < 0.0 ? 0.0 : (tt > (double)(HH - 1) ? (double)(HH - 1) : tt);
    const int rho_i = (int)rr;
    const int th_i  = (int)tt;
    srcbase[t] = (rho_i * WW + th_i) * CC;  // flat element offset of channel 0
  }
  __syncthreads();

  // ---- Phase 1: gather 64 pixels x 128 channels into LDS.
  // One wave moves one pixel row per iteration: 32 lanes x 16B = 512B
  // fully-coalesced read per source pixel, async-copied directly to LDS
  // (CDNA5 GLOBAL_LOAD_ASYNC_TO_LDS_B128, GVS addressing: saddr + v_i32).
  const int q  = t & 31;   // float4 column within the 128-channel row
  const int wv = t >> 5;   // wave id 0..7
#pragma unroll
  for (int m = 0; m < PIX / 8; ++m) {
    const int p = m * 8 + wv;  // tile pixel 0..63
    const unsigned goff =
        (unsigned)(srcbase[p] + c0 + q * 4) * (unsigned)sizeof(float);
    const unsigned lds_off =
        (unsigned)(uintptr_t)&tile[p * LDS_STRIDE + q * 4];
#if defined(__gfx1250__)
    asm volatile("global_load_async_to_lds_b128 %0, %1, %2"
                 :
                 : "v"(lds_off), "v"(goff), "s"(in)
                 : "memory");
#else
    const float4 v = *(const float4*)(in + (size_t)srcbase[p] + c0 + q * 4);
    *(float4*)&tile[p * LDS_STRIDE + q * 4] = v;
    (void)goff; (void)lds_off;
#endif
  }
#if defined(__gfx1250__)
#if __has_builtin(__builtin_amdgcn_s_wait_asynccnt)
  __builtin_amdgcn_s_wait_asynccnt(0);
#else
  asm volatile("s_wait_asynccnt 0" ::: "memory");
#endif
#endif
  __syncthreads();

  // ---- Phase 2: transposed write-out, 128B-contiguous stores.
  // out[(c0+cl)*H*W + i*W + j0+jl] = tile[jl][cl] * wsum
  const float wsum = (float)(0.2989 + 0.587 + 0.114);  // exact numpy f64 sum -> f32
  const size_t outrow = (size_t)i * WW + j0;
#pragma unroll 4
  for (int n = 0; n < (PIX * CHT) / 256; ++n) {  // 32 iterations
    const int lin = n * 256 + t;
    const int jl = lin & (PIX - 1);
    const int cl = lin >> 6;  // log2(PIX)
    const float v = tile[jl * LDS_STRIDE + cl] * wsum;
    out[(size_t)(c0 + cl) * (HH * WW) + outrow + jl] = v;
  }
}

extern "C" void kernel_launch(void* const* d_in, const int* in_sizes, int n_in,
                              void* d_out, int out_size, void* d_ws, size_t ws_size,
                              hipStream_t stream) {
  (void)in_sizes; (void)n_in; (void)out_size; (void)d_ws; (void)ws_size;
  const float* in = (const float*)d_in[0];
  float* out = (float*)d_out;
  dim3 grid(HH, WW / PIX, CC / CHT);  // (256, 4, 4)
  dim3 block(256, 1, 1);
  InverseLogPolarLayer_46909632807255_kernel<<<grid, block, 0, stream>>>(in, out);
}